// Decoder_35837207118002
// MI455X (gfx1250) — compile-verified
//
#include <hip/hip_runtime.h>
#include <math.h>

typedef float v2f __attribute__((ext_vector_type(2)));
typedef float v8f __attribute__((ext_vector_type(8)));

#define NPRE   8192
#define NCUR   16384
#define KNN    8
#define SPLIT  8                    // stream-dimension split factor
#define SLICE1 (NPRE / SPLIT)       // 1024 pre pts  per kernel-1 block (16 KB LDS)
#define SLICE2 (NCUR / SPLIT)       // 2048 cur pts  per kernel-2 block (32 KB LDS)
#define COLS_PER_BLOCK 128          // 8 waves x 16 columns
#define LDS_PAD 32                  // 2 extra tiles so depth-2 prefetch stays in-bounds

// ---------------------------------------------------------------------------
// Pack kernel: WMMA-ready operands.
//   A-form (streamed rows): (-2x, -2y, -2z, ||p||^2)
//   B-form (fixed cols)   : (  x,   y,   z,  1     )
//   N     : ||p||^2  (broadcast into the C accumulator per column)
// sqdist(m,n) = A_m . B_n + C_n exactly, one V_WMMA_F32_16X16X4_F32 per tile.
// ---------------------------------------------------------------------------
__global__ __launch_bounds__(256) void pack_kernel(
    const float* __restrict__ pre, const float* __restrict__ cur,
    float4* __restrict__ preA, float4* __restrict__ preB, float* __restrict__ preN,
    float4* __restrict__ curA, float4* __restrict__ curB, float* __restrict__ curN) {
  int t = blockIdx.x * blockDim.x + threadIdx.x;
  if (t < NCUR) {
    float x = cur[t], y = cur[NCUR + t], z = cur[2 * NCUR + t];
    float n2 = fmaf(x, x, fmaf(y, y, z * z));
    curA[t] = make_float4(-2.0f * x, -2.0f * y, -2.0f * z, n2);
    curB[t] = make_float4(x, y, z, 1.0f);
    curN[t] = n2;
  }
  if (t < NPRE) {
    float x = pre[t], y = pre[NPRE + t], z = pre[2 * NPRE + t];
    float n2 = fmaf(x, x, fmaf(y, y, z * z));
    preA[t] = make_float4(-2.0f * x, -2.0f * y, -2.0f * z, n2);
    preB[t] = make_float4(x, y, z, 1.0f);
    preN[t] = n2;
  }
}

// parallel min-tree over the 8 accumulator values (lowers to v_min3_f32)
__device__ __forceinline__ float min8(v8f d) {
  return fminf(fminf(fminf(d[0], d[1]), fminf(d[2], d[3])),
               fminf(fminf(d[4], d[5]), fminf(d[6], d[7])));
}

// register-resident ascending 8-list bubble insert (fully unrolled)
__device__ __forceinline__ void insert8(float (&bd)[KNN], int (&bi)[KNN],
                                        float dd, int idx) {
#pragma unroll
  for (int k = 0; k < KNN; ++k) {
    if (dd < bd[k]) {
      float td = bd[k]; int ti = bi[k];
      bd[k] = dd; bi[k] = idx;
      dd = td; idx = ti;
    }
  }
}

// ---------------------------------------------------------------------------
// Kernel 1: per-slice argmin over pre for 16 cur columns per wave.
// Block = 8 waves (128 cur columns) x one pre-slice staged in LDS.
// Inner loop: depth-2 software-pipelined ds_load_b64 feeding WMMA.
// ---------------------------------------------------------------------------
__global__ __launch_bounds__(256) void cur2pre_partial(
    const float4* __restrict__ preA, const float4* __restrict__ curB,
    const float* __restrict__ curN, float* __restrict__ pd, int* __restrict__ pi) {
  __shared__ float4 tile[SLICE1 + LDS_PAD];       // 16.5 KB
  const int slice = blockIdx.x % SPLIT;
  const int grp   = blockIdx.x / SPLIT;

  const float4* src = preA + slice * SLICE1;
  for (int i = threadIdx.x; i < SLICE1; i += 256) tile[i] = src[i];
  __syncthreads();
  const v2f* ldsA = (const v2f*)tile;

  const int lane = threadIdx.x & 31;
  const int wave = threadIdx.x >> 5;
  const int n    = lane & 15;
  const int hi   = lane >> 4;                     // 0 / 1 lane half
  const int col  = grp * COLS_PER_BLOCK + wave * 16 + n;   // cur point

  v2f b = ((const v2f*)curB)[col * 2 + hi];       // (x,y) or (z,1)
  float cn = curN[col];
  v8f c;
#pragma unroll
  for (int i = 0; i < 8; ++i) c[i] = cn;

  float bestD = 3.0e38f;
  int   bestI = 0;
  const int idxBase0 = slice * SLICE1 + hi * 8;
  const int laneOff  = n * 2 + hi;                // v2f units

  v2f a0 = ldsA[laneOff];
  v2f a1 = ldsA[32 + laneOff];

  for (int t = 0; t < SLICE1 / 16; t += 2) {
    // prefetch tiles t+2, t+3 (padding covers the final overshoot)
    v2f p0 = ldsA[(t + 2) * 32 + laneOff];
    v2f p1 = ldsA[(t + 3) * 32 + laneOff];

    v8f d = __builtin_amdgcn_wmma_f32_16x16x4_f32(
        false, a0, false, b, (short)0, c, false, false);
    float tmin = min8(d);
    if (tmin < bestD) {                           // rare after warm-up
      bestD = tmin;
      const int base = idxBase0 + t * 16;
#pragma unroll
      for (int i = KNN - 1; i >= 0; --i)          // lowest matching index wins
        if (d[i] == tmin) bestI = base + i;
    }

    d = __builtin_amdgcn_wmma_f32_16x16x4_f32(
        false, a1, false, b, (short)0, c, false, false);
    tmin = min8(d);
    if (tmin < bestD) {
      bestD = tmin;
      const int base = idxBase0 + (t + 1) * 16;
#pragma unroll
      for (int i = KNN - 1; i >= 0; --i)
        if (d[i] == tmin) bestI = base + i;
    }

    a0 = p0;
    a1 = p1;
  }

  float oD = __shfl_xor(bestD, 16, 32);
  int   oI = __shfl_xor(bestI, 16, 32);
  if (oD < bestD || (oD == bestD && oI < bestI)) { bestD = oD; bestI = oI; }
  if (lane < 16) {
    pd[slice * NCUR + col] = bestD;
    pi[slice * NCUR + col] = bestI;
  }
}

// Merge the SPLIT per-slice argmins (slices ordered by pre-index -> '<' keeps
// the lowest index on ties).
__global__ __launch_bounds__(256) void c2p_merge(
    const float* __restrict__ pd, const int* __restrict__ pi,
    int* __restrict__ c2p) {
  int j = blockIdx.x * 256 + threadIdx.x;
  float best = 3.0e38f;
  int   bi   = 0;
#pragma unroll
  for (int s = 0; s < SPLIT; ++s) {
    float d = pd[s * NCUR + j];
    int   i = pi[s * NCUR + j];
    if (d < best) { best = d; bi = i; }
  }
  c2p[j] = bi;
}

// ---------------------------------------------------------------------------
// Kernel 2: per-slice top-8 nearest cur points for 16 pre columns per wave.
// Block = 8 waves (128 pre columns) x one cur-slice staged in LDS.
// ---------------------------------------------------------------------------
__global__ __launch_bounds__(256) void knn_partial(
    const float4* __restrict__ curA, const float4* __restrict__ preB,
    const float* __restrict__ preN, float* __restrict__ kd, int* __restrict__ ki) {
  __shared__ float4 tile[SLICE2 + LDS_PAD];       // 32.5 KB
  const int slice = blockIdx.x % SPLIT;
  const int grp   = blockIdx.x / SPLIT;

  const float4* src = curA + slice * SLICE2;
  for (int i = threadIdx.x; i < SLICE2; i += 256) tile[i] = src[i];
  __syncthreads();
  const v2f* ldsA = (const v2f*)tile;

  const int lane = threadIdx.x & 31;
  const int wave = threadIdx.x >> 5;
  const int n    = lane & 15;
  const int hi   = lane >> 4;
  const int col  = grp * COLS_PER_BLOCK + wave * 16 + n;   // pre point

  v2f b = ((const v2f*)preB)[col * 2 + hi];
  float pn = preN[col];
  v8f c;
#pragma unroll
  for (int i = 0; i < 8; ++i) c[i] = pn;

  float bd[KNN];
  int   bi[KNN];
#pragma unroll
  for (int k = 0; k < KNN; ++k) { bd[k] = 3.0e38f; bi[k] = 0; }

  const int idxBase0 = slice * SLICE2 + hi * 8;
  const int laneOff  = n * 2 + hi;

  v2f a0 = ldsA[laneOff];
  v2f a1 = ldsA[32 + laneOff];

  for (int t = 0; t < SLICE2 / 16; t += 2) {
    v2f p0 = ldsA[(t + 2) * 32 + laneOff];
    v2f p1 = ldsA[(t + 3) * 32 + laneOff];

    v8f d = __builtin_amdgcn_wmma_f32_16x16x4_f32(
        false, a0, false, b, (short)0, c, false, false);
    float tmin = min8(d);
    if (tmin < bd[KNN - 1]) {                     // tile-level guard
      const int base = idxBase0 + t * 16;
#pragma unroll
      for (int i = 0; i < 8; ++i) {
        float dd = d[i];
        if (dd < bd[KNN - 1]) insert8(bd, bi, dd, base + i);
      }
    }

    d = __builtin_amdgcn_wmma_f32_16x16x4_f32(
        false, a1, false, b, (short)0, c, false, false);
    tmin = min8(d);
    if (tmin < bd[KNN - 1]) {
      const int base = idxBase0 + (t + 1) * 16;
#pragma unroll
      for (int i = 0; i < 8; ++i) {
        float dd = d[i];
        if (dd < bd[KNN - 1]) insert8(bd, bi, dd, base + i);
      }
    }

    a0 = p0;
    a1 = p1;
  }

  // snapshot-exchange with the partner lane half, then merge
  float od[KNN];
  int   oi[KNN];
#pragma unroll
  for (int k = 0; k < KNN; ++k) {
    od[k] = __shfl_xor(bd[k], 16, 32);
    oi[k] = __shfl_xor(bi[k], 16, 32);
  }
#pragma unroll
  for (int k = 0; k < KNN; ++k)
    if (od[k] < bd[KNN - 1]) insert8(bd, bi, od[k], oi[k]);

  if (lane < 16) {
    const int base = (slice * NPRE + col) * KNN;
#pragma unroll
    for (int k = 0; k < KNN; ++k) { kd[base + k] = bd[k]; ki[base + k] = bi[k]; }
  }
}

// Merge SPLIT sorted 8-lists per pre point, apply cur2pre mask, emit output.
__global__ __launch_bounds__(256) void knn_final(
    const float* __restrict__ kd, const int* __restrict__ ki,
    const int* __restrict__ c2p, const float* __restrict__ ups,
    float* __restrict__ out) {
  int p = blockIdx.x * 256 + threadIdx.x;         // < NPRE
  float bd[KNN];
  int   bi[KNN];
#pragma unroll
  for (int k = 0; k < KNN; ++k) { bd[k] = 3.0e38f; bi[k] = 0; }

  for (int s = 0; s < SPLIT; ++s) {
    const int base = (s * NPRE + p) * KNN;
#pragma unroll
    for (int k = 0; k < KNN; ++k) {
      float dd = kd[base + k];
      if (dd < bd[KNN - 1]) insert8(bd, bi, dd, ki[base + k]);
    }
  }

  float s = 0.0f;
#pragma unroll
  for (int k = 0; k < KNN; ++k) {
    float m = (c2p[bi[k]] == p) ? 1.0f : 0.0f;
    s += m * sqrtf(fmaxf(bd[k], 0.0f));
  }
  out[p] = s / ups[p];
}

// ---------------------------------------------------------------------------
// Workspace layout (bytes, 256-aligned):
//   curA 262144 @ 0        curB 262144 @ 262144
//   preA 131072 @ 524288   preB 131072 @ 655360
//   curN  65536 @ 786432   preN  32768 @ 851968
//   c2p   65536 @ 884736
//   pd   524288 @ 950272   pi   524288 @ 1474560   (SPLIT x NCUR)
//   kd  2097152 @ 1998848  ki  2097152 @ 4096000   (SPLIT x NPRE x 8)
//   total ~6.0 MB
// ---------------------------------------------------------------------------
extern "C" void kernel_launch(void* const* d_in, const int* in_sizes, int n_in,
                              void* d_out, int out_size, void* d_ws, size_t ws_size,
                              hipStream_t stream) {
  const float* pre = (const float*)d_in[0];   // (1, 3, 8192)
  const float* cur = (const float*)d_in[1];   // (1, 3, 16384)
  const float* ups = (const float*)d_in[2];   // (1, 8192)
  float* out = (float*)d_out;                 // (1, 8192)

  char* ws = (char*)d_ws;
  float4* curA = (float4*)(ws + 0);
  float4* curB = (float4*)(ws + 262144);
  float4* preA = (float4*)(ws + 524288);
  float4* preB = (float4*)(ws + 655360);
  float*  curN = (float*) (ws + 786432);
  float*  preN = (float*) (ws + 851968);
  int*    c2p  = (int*)   (ws + 884736);
  float*  pd   = (float*) (ws + 950272);
  int*    pi   = (int*)   (ws + 1474560);
  float*  kd   = (float*) (ws + 1998848);
  int*    ki   = (int*)   (ws + 4096000);

  pack_kernel<<<(NCUR + 255) / 256, 256, 0, stream>>>(
      pre, cur, preA, preB, preN, curA, curB, curN);

  // 1) per-slice argmin: (16384/128 col-groups) x 8 slices = 1024 blocks
  cur2pre_partial<<<(NCUR / COLS_PER_BLOCK) * SPLIT, 256, 0, stream>>>(
      preA, curB, curN, pd, pi);
  c2p_merge<<<NCUR / 256, 256, 0, stream>>>(pd, pi, c2p);

  // 2) per-slice top-8: (8192/128 col-groups) x 8 slices = 512 blocks
  knn_partial<<<(NPRE / COLS_PER_BLOCK) * SPLIT, 256, 0, stream>>>(
      curA, preB, preN, kd, ki);
  knn_final<<<NPRE / 256, 256, 0, stream>>>(kd, ki, c2p, ups, out);
}